// DiffNNSearch_65111704207552
// MI455X (gfx1250) — compile-verified
//
#include <hip/hip_runtime.h>

typedef float v2f __attribute__((ext_vector_type(2)));
typedef float v4f __attribute__((ext_vector_type(4)));
typedef float v8f __attribute__((ext_vector_type(8)));

#define B_    2
#define N_    8192
#define D_    128
#define TILES (N_ / 16)        // 512 column tiles of 16
#define WAVES 4                // waves per block (wave32)
#define ROWS_PER_BLOCK (16 * WAVES)
#define LDS_STRIDE 132         // 128 + 4 floats pad -> conflict-free b64 fragment reads

// ---------------------------------------------------------------------------
// Kernel 1: zero-fill the one-hot output (512 MB streamed stores).
// ---------------------------------------------------------------------------
__global__ void zero_fill_kernel(float* __restrict__ out, size_t n4) {
    size_t i      = (size_t)blockIdx.x * blockDim.x + threadIdx.x;
    size_t stride = (size_t)gridDim.x * blockDim.x;
    v4f z = {0.0f, 0.0f, 0.0f, 0.0f};
    v4f* p = (v4f*)out;
    for (; i < n4; i += stride) {
        __builtin_nontemporal_store(z, p + i);   // don't pollute L2
    }
}

// ---------------------------------------------------------------------------
// Kernel 2: y2[b,m] = ||feats1[b,m,:]||^2   (16384 values, trivial)
// ---------------------------------------------------------------------------
__global__ void y2_kernel(const float* __restrict__ Y, float* __restrict__ y2) {
    int i = blockIdx.x * blockDim.x + threadIdx.x;
    if (i >= B_ * N_) return;
    const v4f* p = (const v4f*)(Y + (size_t)i * D_);
    float s = 0.0f;
#pragma unroll
    for (int k = 0; k < D_ / 4; ++k) {
        v4f v = p[k];
        s += v.x * v.x + v.y * v.y + v.z * v.z + v.w * v.w;
    }
    y2[i] = s;
}

// ---------------------------------------------------------------------------
// Kernel 3: WMMA nearest-neighbor argmin.
//
// Block = 64 rows of feats0 (4 waves x private 16x128 A tile in 64 VGPRs).
// All waves walk all 512 column tiles; each 16x128 Y tile is staged into LDS
// once per block (4x L2-traffic reduction) with software double-buffering:
//   iter t: issue global loads for tile t+1 -> regs
//           bulk ds_load of ALL 32 B fragments -> regs  (DS pipelined, one
//                                                        exposed latency/tile)
//           4 independent WMMA accumulation chains on tile t
//           store regs -> ybuf[(t+1)&1]; __syncthreads()
//
// fp32 16x16x4 fragment layouts (ISA 7.12.2), lane l, l15=l&15, hi=l>>4:
//   A (16x4): {A[l15, 2hi], A[l15, 2hi+1]}  -> consecutive float2
//   B (4x16): {B[2hi, l15], B[2hi+1, l15]}  -> row-major y -> consecutive float2
//   C/D     : vgpr i = S[i + 8*hi, l15]     -> each lane owns ONE column/tile
// LDS stride 132 floats: fragment bank = (4*l15 + 4k + 2hi) mod 64, all 32
// lanes hit disjoint bank pairs -> conflict-free ds_load_b64.
// ---------------------------------------------------------------------------
__global__ __launch_bounds__(32 * WAVES) void nn_argmin_kernel(
    const float* __restrict__ X0, const float* __restrict__ X1,
    const float* __restrict__ y2g, float* __restrict__ asgn,
    int* __restrict__ idxout) {

    __shared__ float ybuf[2][16 * LDS_STRIDE];

    const int blk  = blockIdx.x;
    const int b    = blk / (N_ / ROWS_PER_BLOCK);
    const int row0 = (blk % (N_ / ROWS_PER_BLOCK)) * ROWS_PER_BLOCK;
    const int tid  = threadIdx.x;
    const int wave = tid >> 5;
    const int lane = tid & 31;
    const int l15  = lane & 15;
    const int hi   = lane >> 4;

    const float* X  = X0  + ((size_t)b * N_ + row0 + 16 * wave) * D_;
    const float* Y  = X1  + (size_t)b * N_ * D_;
    const float* y2 = y2g + (size_t)b * N_;

    // Private A tile: 16 rows x 128 K as 32 K-slices of 16x4 (64 VGPRs),
    // reused across all 512 column tiles.
    v2f a[32];
#pragma unroll
    for (int k = 0; k < 32; ++k)
        a[k] = *(const v2f*)(X + (size_t)l15 * D_ + 4 * k + 2 * hi);

    // Cooperative staging: 8 KB tile = 512 float4; 128 threads x 4 float4.
    v4f stage[4];
    {
#pragma unroll
        for (int j = 0; j < 4; ++j) {
            int e = j * 128 + tid, r = e >> 5, c4 = e & 31;
            stage[j] = *(const v4f*)(Y + (size_t)r * D_ + c4 * 4);
        }
#pragma unroll
        for (int j = 0; j < 4; ++j) {
            int e = j * 128 + tid, r = e >> 5, c4 = e & 31;
            *(v4f*)(&ybuf[0][r * LDS_STRIDE + c4 * 4]) = stage[j];
        }
    }
    __syncthreads();

    float best[8];
    int   bidx[8];
#pragma unroll
    for (int i = 0; i < 8; ++i) { best[i] = 3.402823466e38f; bidx[i] = 0x7fffffff; }

    for (int t = 0; t < TILES; ++t) {
        const int cur = t & 1;

        // Issue global loads for tile t+1 (latency covered by the WMMA work).
        if (t + 1 < TILES) {
            const float* Yn = Y + (size_t)(t + 1) * 16 * D_;
#pragma unroll
            for (int j = 0; j < 4; ++j) {
                int e = j * 128 + tid, r = e >> 5, c4 = e & 31;
                stage[j] = *(const v4f*)(Yn + (size_t)r * D_ + c4 * 4);
            }
        }

        // Bulk-load ALL 32 B fragments first: the ds_load_2addr_b64s issue
        // back-to-back and the WMMA run drains them with staged dscnt waits,
        // exposing only one LDS latency per tile (vs. one per fragment pair).
        const float* yl = &ybuf[cur][l15 * LDS_STRIDE + 2 * hi];
        v2f bf[32];
#pragma unroll
        for (int k = 0; k < 32; ++k)
            bf[k] = *(const v2f*)(yl + 4 * k);

        // Four independent accumulation chains: dependent-WMMA latency on the
        // C operand can never bound the issue rate.
        v8f c0 = {}, c1 = {}, c2 = {}, c3 = {};
#pragma unroll
        for (int k = 0; k < 8; ++k) {
            c0 = __builtin_amdgcn_wmma_f32_16x16x4_f32(false, a[4 * k],     false, bf[4 * k],
                                                       (short)0, c0, false, false);
            c1 = __builtin_amdgcn_wmma_f32_16x16x4_f32(false, a[4 * k + 1], false, bf[4 * k + 1],
                                                       (short)0, c1, false, false);
            c2 = __builtin_amdgcn_wmma_f32_16x16x4_f32(false, a[4 * k + 2], false, bf[4 * k + 2],
                                                       (short)0, c2, false, false);
            c3 = __builtin_amdgcn_wmma_f32_16x16x4_f32(false, a[4 * k + 3], false, bf[4 * k + 3],
                                                       (short)0, c3, false, false);
        }

        const int   m0  = t * 16;
        const float yy  = y2[m0 + l15];
        const int   col = m0 + l15;
#pragma unroll
        for (int i = 0; i < 8; ++i) {
            float s = yy - 2.0f * ((c0[i] + c1[i]) + (c2[i] + c3[i]));
            if (s < best[i]) { best[i] = s; bidx[i] = col; }  // ascending col: first-on-tie
        }

        // Publish tile t+1 into the alternate buffer.
        if (t + 1 < TILES) {
            float* dst = ybuf[cur ^ 1];
#pragma unroll
            for (int j = 0; j < 4; ++j) {
                int e = j * 128 + tid, r = e >> 5, c4 = e & 31;
                *(v4f*)(&dst[r * LDS_STRIDE + c4 * 4]) = stage[j];
            }
        }
        __syncthreads();
    }

    // Reduce across the 16 lanes of each half (xor masks < 16 stay in-half),
    // tie-break toward the smaller column index (jnp.argmax first-max rule).
#pragma unroll
    for (int off = 1; off < 16; off <<= 1) {
#pragma unroll
        for (int i = 0; i < 8; ++i) {
            float os = __shfl_xor(best[i], off, 32);
            int   oi = __shfl_xor(bidx[i], off, 32);
            if (os < best[i] || (os == best[i] && oi < bidx[i])) {
                best[i] = os; bidx[i] = oi;
            }
        }
    }

    // Each wave owns 16 distinct rows -> no cross-wave combine needed.
    if (l15 == 0) {
#pragma unroll
        for (int i = 0; i < 8; ++i) {
            const size_t grow = (size_t)b * N_ + row0 + 16 * wave + i + 8 * hi;
            idxout[grow]                      = bidx[i];
            asgn[grow * (size_t)N_ + bidx[i]] = 1.0f;
        }
    }
}

// ---------------------------------------------------------------------------
extern "C" void kernel_launch(void* const* d_in, const int* in_sizes, int n_in,
                              void* d_out, int out_size, void* d_ws, size_t ws_size,
                              hipStream_t stream) {
    const float* feats0 = (const float*)d_in[0];   // [B, N, D] fp32
    const float* feats1 = (const float*)d_in[1];   // [B, N, D] fp32
    // d_in[2] (temp) unused: argmax(softmax(-d/t)) invariant for t > 0.

    float* asgn   = (float*)d_out;                                  // [B, N, N] fp32
    int*   idxout = (int*)((float*)d_out + (size_t)B_ * N_ * N_);   // [B, N] int32
    float* y2     = (float*)d_ws;                                   // [B, N] scratch

    // 1) Zero the one-hot output (512 MB streamed stores).
    const size_t n4 = (size_t)B_ * N_ * N_ / 4;
    zero_fill_kernel<<<8192, 256, 0, stream>>>(asgn, n4);

    // 2) ||y||^2 per reference point.
    y2_kernel<<<(B_ * N_ + 255) / 256, 256, 0, stream>>>(feats1, y2);

    // 3) WMMA argmin + scatter the 1.0s and indices.
    nn_argmin_kernel<<<B_ * (N_ / ROWS_PER_BLOCK), 32 * WAVES, 0, stream>>>(
        feats0, feats1, y2, asgn, idxout);
}